// RankLoss_55250459296257
// MI455X (gfx1250) — compile-verified
//
#include <hip/hip_runtime.h>
#include <hip/hip_bf16.h>

// ---------------------------------------------------------------------------
// RankLoss on MI455X (gfx1250):
//   scores = dat(8192x4096 f32) @ w(4096 f32)   -> WMMA f32 16x16x4 path
//   loss   = sum_{i pos, j neg} max(s_j - s_i', 0) / (npos*nneg), s_i' = s_i - 0.1
// Bandwidth-bound: 134MB of dat @ 23.3 TB/s ~= 5.8us floor. WMMA's redundant
// columns (w replicated across all 16 B columns) are free under that bound.
// This revision: uniform-trip-count inner loop (no exec-mask churn) and 8
// independent loads + 8 WMMAs per iteration for memory-level parallelism.
// ---------------------------------------------------------------------------

#define RL_N 8192
#define RL_D 4096
#define RL_MARGIN 0.1f
#define RL_KCHUNK (RL_D / 8)   // 512 columns per wave
#define RL_KSTEP 32            // 8 WMMAs (K=4 each) per outer iteration

typedef __attribute__((ext_vector_type(2))) float v2f;
typedef __attribute__((ext_vector_type(8))) float v8f;

union V8FU { v8f v; float f[8]; };

// ---------------------------------------------------------------------------
// Kernel 0: zero the global loss accumulator.
// ---------------------------------------------------------------------------
__global__ void rl_init_kernel(float* accum) {
    if (threadIdx.x == 0 && blockIdx.x == 0) accum[0] = 0.0f;
}

// ---------------------------------------------------------------------------
// Kernel 1: scores[row] = dot(dat[row,:], w) via V_WMMA_F32_16X16X4_F32.
// Block = 256 threads = 8 waves (wave32). Block owns rows [16*bid, 16*bid+16).
// Wave wv covers columns [wv*512, wv*512+512) in groups of 8 K=4 WMMA steps.
//
// A (16x4 f32) VGPR layout (ISA 7.12.2): lanes 0-15 -> M=lane, K={0,1} in
// {v0,v1}; lanes 16-31 -> M=lane-16, K={2,3}.  B (4x16) mirrors in K across
// the lane halves; since we replicate w over all 16 B columns, each lane just
// needs w[k], w[k+1] at the same k offsets as its A fragment.
// C/D (16x16 f32): lane 0 vgprs 0..7 = rows 0..7 (col 0), lane 16 = rows 8..15.
// ---------------------------------------------------------------------------
__global__ __launch_bounds__(256)
void rl_matvec_wmma_kernel(const float* __restrict__ dat,
                           const float* __restrict__ w,
                           float* __restrict__ scores) {
    __shared__ float wlds[RL_D];          // 16 KB: whole w vector
    __shared__ float partial[8 * 16];     // per-wave partial score rows

    const int tid = threadIdx.x;
    // Stage w into LDS once (coalesced b64 loads, 2 floats/thread/iter).
    for (int i = tid * 2; i < RL_D; i += 256 * 2) {
        *(v2f*)&wlds[i] = *(const v2f*)&w[i];
    }
    __syncthreads();

    const int wave  = tid >> 5;           // 0..7
    const int lane  = tid & 31;
    const int m     = lane & 15;          // row within tile
    const int khalf = lane >> 4;          // 0 or 1 -> K offset 0 or 2
    const int row0  = blockIdx.x * 16;

    // Fold all per-lane offsets into base pointers so the loop counter is
    // wave-uniform (scalar branch, no per-iteration exec masking).
    const float* __restrict__ aptr =
        dat + (size_t)(row0 + m) * RL_D + wave * RL_KCHUNK + 2 * khalf;
    const float* __restrict__ bptr = &wlds[wave * RL_KCHUNK + 2 * khalf];

    v8f c = {};
#pragma unroll 1
    for (int kk = 0; kk < RL_KCHUNK; kk += RL_KSTEP) {
        // Prefetch one iteration ahead (gfx1250 global_prefetch_b8;
        // speculative, OOB translations silently dropped).
        __builtin_prefetch(aptr + kk + RL_KSTEP, 0, 1);

        v2f a[8];
        v2f b[8];
#pragma unroll
        for (int j = 0; j < 8; ++j) {        // 8 independent global b64 loads
            a[j] = *(const v2f*)(aptr + kk + 4 * j);
        }
#pragma unroll
        for (int j = 0; j < 8; ++j) {        // 8 independent ds b64 loads
            b[j] = *(const v2f*)(bptr + kk + 4 * j);
        }
#pragma unroll
        for (int j = 0; j < 8; ++j) {        // 8 chained WMMA accumulates
            c = __builtin_amdgcn_wmma_f32_16x16x4_f32(
                    /*neg_a=*/false, a[j], /*neg_b=*/false, b[j],
                    /*c_mod=*/(short)0, c, /*reuse_a=*/false, /*reuse_b=*/false);
        }
    }

    // Every C column is identical (B columns replicated). Lane 0 holds rows
    // 0..7 in c[0..7]; lane 16 holds rows 8..15.
    if ((lane & 15) == 0) {
        V8FU u; u.v = c;
        const int rbase = khalf * 8;
#pragma unroll
        for (int r = 0; r < 8; ++r) partial[wave * 16 + rbase + r] = u.f[r];
    }
    __syncthreads();

    // Reduce the 8 per-wave partials for each of the 16 rows.
    if (tid < 16) {
        float s = 0.0f;
#pragma unroll
        for (int wv = 0; wv < 8; ++wv) s += partial[wv * 16 + tid];
        scores[row0 + tid] = s;
    }
}

// ---------------------------------------------------------------------------
// Kernel 2: pairwise hinge. Grid (32 i-blocks, 8 j-chunks). Each block caches
// a 1024-entry j-chunk of adjusted scores + neg mask in LDS; thread owns one
// positive i; wave-shuffle reduce; one f32 atomic per wave.
// Adjusted score s' = s - MARGIN*(label==1) is correct for BOTH roles:
// positives use s-MARGIN, negatives are unadjusted.
// ---------------------------------------------------------------------------
#define RL_JCHUNK 1024

__global__ __launch_bounds__(256)
void rl_pair_hinge_kernel(const float* __restrict__ scores,
                          const int* __restrict__ labels,
                          float* __restrict__ accum) {
    __shared__ float sj[RL_JCHUNK];
    __shared__ float nmask[RL_JCHUNK];

    const int tid = threadIdx.x;
    const int j0  = blockIdx.y * RL_JCHUNK;

    for (int t = tid; t < RL_JCHUNK; t += 256) {
        const int   lab = labels[j0 + t];
        const float s   = scores[j0 + t];
        sj[t]    = s - RL_MARGIN * (float)(lab == 1);
        nmask[t] = (lab == 0) ? 1.0f : 0.0f;
    }
    __syncthreads();

    const int   i    = blockIdx.x * 256 + tid;
    const int   labi = labels[i];
    const float si   = scores[i] - RL_MARGIN * (float)(labi == 1);

    float acc = 0.0f;
    if (labi == 1) {
#pragma unroll 8
        for (int t = 0; t < RL_JCHUNK; ++t) {
            acc += fmaxf(sj[t] - si, 0.0f) * nmask[t];
        }
    }

    // wave32 shuffle reduction
#pragma unroll
    for (int off = 16; off > 0; off >>= 1) acc += __shfl_down(acc, off, 32);
    if ((tid & 31) == 0) atomicAdd(accum, acc);
}

// ---------------------------------------------------------------------------
// Kernel 3: count positives, produce final scalar loss.
// ---------------------------------------------------------------------------
__global__ __launch_bounds__(256)
void rl_finalize_kernel(const int* __restrict__ labels,
                        const float* __restrict__ accum,
                        float* __restrict__ out, int n) {
    __shared__ int wsum[8];
    const int tid = threadIdx.x;

    int npos = 0;
    for (int idx = tid; idx < n; idx += 256) npos += (labels[idx] == 1) ? 1 : 0;
#pragma unroll
    for (int off = 16; off > 0; off >>= 1) npos += __shfl_down(npos, off, 32);
    if ((tid & 31) == 0) wsum[tid >> 5] = npos;
    __syncthreads();

    if (tid == 0) {
        int p = 0;
#pragma unroll
        for (int wv = 0; wv < 8; ++wv) p += wsum[wv];
        const float npairs = (float)p * (float)(n - p);
        const float total  = accum[0];
        out[0] = (npairs > 0.0f) ? (total / fmaxf(npairs, 1.0f)) : 0.0f;
    }
}

// ---------------------------------------------------------------------------
extern "C" void kernel_launch(void* const* d_in, const int* in_sizes, int n_in,
                              void* d_out, int out_size, void* d_ws, size_t ws_size,
                              hipStream_t stream) {
    const float* dat    = (const float*)d_in[0];   // [N, D] f32
    const int*   labels = (const int*)d_in[1];     // [N] i32
    const float* w      = (const float*)d_in[2];   // [D] f32
    float*       out    = (float*)d_out;           // [1] f32

    float* scores = (float*)d_ws;                  // N floats
    float* accum  = scores + RL_N;                 // 1 float

    rl_init_kernel<<<1, 1, 0, stream>>>(accum);
    rl_matvec_wmma_kernel<<<RL_N / 16, 256, 0, stream>>>(dat, w, scores);
    rl_pair_hinge_kernel<<<dim3(RL_N / 256, RL_N / RL_JCHUNK), 256, 0, stream>>>(
        scores, labels, accum);
    rl_finalize_kernel<<<1, 256, 0, stream>>>(labels, accum, out, RL_N);
}